// SpGAT_13374528160102
// MI455X (gfx1250) — compile-verified
//
#include <hip/hip_runtime.h>
#include <hip/hip_bf16.h>

typedef float v2f __attribute__((ext_vector_type(2)));
typedef float v8f __attribute__((ext_vector_type(8)));

#define N_NODES   50000
#define N_EDGES   1600000
#define INIT_DIM  256
#define HID_DIM   64
#define NHEADS    8
#define CAT_DIM   512   /* NHEADS * HID_DIM */
#define EMBED_DIM 128
#define LRELU_ALPHA 0.2f
#define EPS 1e-16f

__device__ __forceinline__ float lrelu(float x) { return x >= 0.f ? x : LRELU_ALPHA * x; }
__device__ __forceinline__ float elu(float x)   { return x > 0.f ? x : __expf(x) - 1.f; }

__global__ __launch_bounds__(256) void k_zero(float* __restrict__ p, int n) {
    int i = blockIdx.x * 256 + threadIdx.x;
    if (i < n) p[i] = 0.f;
}

// ---------------------------------------------------------------------------
// f32 WMMA GEMM: C[M,N] = A[M,K] * B(K,N)
// MODE 0: B(k,c) = W_in[(c>>6)*(K*64) + k*64 + (c&63)]   (head-blocked [8][K][64])
// MODE 1: B(k,c) = B[k*N + c]                            (plain row-major)
// Block = 256 threads = 8 waves; block covers 16 rows x 128 cols;
// wave w covers the 16x16 tile at column (blockCol*128 + w*16).
// A panel (16x64) staged in LDS with stride 66 (bank-conflict-free reads).
// ---------------------------------------------------------------------------
template <int MODE>
__global__ __launch_bounds__(256) void k_wmma_gemm(const float* __restrict__ A,
                                                   const float* __restrict__ B,
                                                   float* __restrict__ C,
                                                   int K, int N) {
    __shared__ float Alds[16 * 66];
    const int lane  = threadIdx.x & 31;
    const int wave  = threadIdx.x >> 5;
    const int colBlocks = N >> 7;               // N / 128
    const int m0 = (blockIdx.x / colBlocks) * 16;
    const int n0 = (blockIdx.x % colBlocks) * 128 + wave * 16;

    const int mrow  = lane & 15;                 // M (for A) / N (for B) within tile
    const int khalf = (lane >> 4) << 1;          // lanes 0-15 -> K{0,1}; 16-31 -> K{2,3}

    v8f acc = {0.f, 0.f, 0.f, 0.f, 0.f, 0.f, 0.f, 0.f};

    for (int k0 = 0; k0 < K; k0 += 64) {
        // cooperative A-panel load: 16 rows x 64 k  (float4 per thread, coalesced)
        {
            const int r = threadIdx.x >> 4;
            const int s = (threadIdx.x & 15) << 2;
            const float4 v = *(const float4*)(A + (size_t)(m0 + r) * K + k0 + s);
            Alds[r * 66 + s + 0] = v.x;
            Alds[r * 66 + s + 1] = v.y;
            Alds[r * 66 + s + 2] = v.z;
            Alds[r * 66 + s + 3] = v.w;
        }
        __syncthreads();
#pragma unroll
        for (int kk = 0; kk < 64; kk += 4) {
            v2f a, b;
            const int ka = kk + khalf;
            a.x = Alds[mrow * 66 + ka + 0];
            a.y = Alds[mrow * 66 + ka + 1];
            const int kb = k0 + kk + khalf;
            const int c  = n0 + mrow;
            if (MODE == 0) {
                const float* bp = B + (size_t)(c >> 6) * ((size_t)K * HID_DIM) + (c & 63);
                b.x = bp[(size_t)(kb + 0) * HID_DIM];
                b.y = bp[(size_t)(kb + 1) * HID_DIM];
            } else {
                b.x = B[(size_t)(kb + 0) * N + c];
                b.y = B[(size_t)(kb + 1) * N + c];
            }
            acc = __builtin_amdgcn_wmma_f32_16x16x4_f32(false, a, false, b,
                                                        (short)0, acc, false, false);
        }
        __syncthreads();
    }
    // C/D layout: VGPR i -> M = i + (lane>=16 ? 8 : 0), N = lane&15
    const int crow = m0 + ((lane >> 4) << 3);
    const int ccol = n0 + mrow;
#pragma unroll
    for (int i = 0; i < 8; ++i)
        C[(size_t)(crow + i) * N + ccol] = acc[i];
}

// ---------------------------------------------------------------------------
// Per-(node,head) attention scalars for layer 1: F = 64.
// as1[n*8+h] = h_all[n, h*64: ] . a_in[h][0:64]; ad1 uses a_in[h][64:128].
// One wave per (node,head); butterfly reduce (wave32).
// ---------------------------------------------------------------------------
__global__ __launch_bounds__(256) void k_node_scalars1(const float* __restrict__ h_all,
                                                       const float* __restrict__ a_in,
                                                       float* __restrict__ as1,
                                                       float* __restrict__ ad1) {
    const int lane = threadIdx.x & 31;
    const int idx  = blockIdx.x * 8 + (threadIdx.x >> 5);   // n*8 + h
    if (idx >= N_NODES * NHEADS) return;
    const int n = idx >> 3, h = idx & 7;
    const float* hp = h_all + (size_t)n * CAT_DIM + h * HID_DIM;
    const float* ap = a_in + h * (2 * HID_DIM);
    const float h0 = hp[lane], h1 = hp[lane + 32];
    float ss = h0 * ap[lane]      + h1 * ap[lane + 32];
    float sd = h0 * ap[64 + lane] + h1 * ap[96 + lane];
#pragma unroll
    for (int o = 16; o > 0; o >>= 1) {
        ss += __shfl_xor(ss, o, 32);
        sd += __shfl_xor(sd, o, 32);
    }
    if (lane == 0) { as1[idx] = ss; ad1[idx] = sd; }
}

// Per-node scalars for layer 2: F = 128, one wave per node.
__global__ __launch_bounds__(256) void k_node_scalars2(const float* __restrict__ h2,
                                                       const float* __restrict__ a_out,
                                                       float* __restrict__ as2,
                                                       float* __restrict__ ad2) {
    const int lane = threadIdx.x & 31;
    const int n = blockIdx.x * 8 + (threadIdx.x >> 5);
    if (n >= N_NODES) return;
    const float* hp = h2 + (size_t)n * EMBED_DIM;
    float ss = 0.f, sd = 0.f;
#pragma unroll
    for (int j = 0; j < 4; ++j) {
        const float hv = hp[lane + 32 * j];
        ss += hv * a_out[lane + 32 * j];
        sd += hv * a_out[128 + lane + 32 * j];
    }
#pragma unroll
    for (int o = 16; o > 0; o >>= 1) {
        ss += __shfl_xor(ss, o, 32);
        sd += __shfl_xor(sd, o, 32);
    }
    if (lane == 0) { as2[n] = ss; ad2[n] = sd; }
}

// ---------------------------------------------------------------------------
// Layer-1 edge scatter: one wave per edge, all 8 heads fused.
// Lanes 0-7 own head weights (+rowsum atomics); all lanes stream the full
// contiguous 512-float h_all[dst] row into atomicAdd(agg1[src]).
// ---------------------------------------------------------------------------
__global__ __launch_bounds__(256) void k_edge1(const int* __restrict__ src,
                                               const int* __restrict__ dst,
                                               const float* __restrict__ h_all,
                                               const float* __restrict__ as1,
                                               const float* __restrict__ ad1,
                                               float* __restrict__ rowsum1,
                                               float* __restrict__ agg1) {
    const int lane = threadIdx.x & 31;
    const int e = blockIdx.x * 8 + (threadIdx.x >> 5);
    if (e >= N_EDGES) return;
    const int s = src[e], d = dst[e];
    float wv = 0.f;
    if (lane < 8) {
        const float lg = as1[s * 8 + lane] + ad1[d * 8 + lane];
        wv = __expf(-lrelu(lg));
        atomicAdd(&rowsum1[s * 8 + lane], wv);
    }
    const float* hp = h_all + (size_t)d * CAT_DIM;
    float* op = agg1 + (size_t)s * CAT_DIM;
#pragma unroll
    for (int i = 0; i < 16; ++i) {                 // 512 floats / 32 lanes
        const int idx = i * 32 + lane;
        const float w = __shfl(wv, i >> 1, 32);    // head = idx>>6 = i>>1 (uniform per i)
        atomicAdd(&op[idx], w * hp[idx]);
    }
}

// Layer-2 edge scatter: one wave per edge, F = 128, single head.
__global__ __launch_bounds__(256) void k_edge2(const int* __restrict__ src,
                                               const int* __restrict__ dst,
                                               const float* __restrict__ h2,
                                               const float* __restrict__ as2,
                                               const float* __restrict__ ad2,
                                               float* __restrict__ rowsum2,
                                               float* __restrict__ agg2,
                                               float* __restrict__ ework) {
    const int lane = threadIdx.x & 31;
    const int e = blockIdx.x * 8 + (threadIdx.x >> 5);
    if (e >= N_EDGES) return;
    const int s = src[e], d = dst[e];
    const float wv = __expf(-lrelu(as2[s] + ad2[d]));  // scalar loads broadcast
    if (lane == 0) {
        atomicAdd(&rowsum2[s], wv);
        ework[e] = wv;
    }
    const float* hp = h2 + (size_t)d * EMBED_DIM;
    float* op = agg2 + (size_t)s * EMBED_DIM;
#pragma unroll
    for (int j = 0; j < 4; ++j) {
        const int idx = j * 32 + lane;
        atomicAdd(&op[idx], wv * hp[idx]);
    }
}

// agg1 <- elu(agg1 / (rowsum1 + eps))   (in place; becomes xcat)
__global__ __launch_bounds__(256) void k_norm_elu1(float* __restrict__ agg1,
                                                   const float* __restrict__ rowsum1) {
    const int i = blockIdx.x * 256 + threadIdx.x;
    if (i >= N_NODES * CAT_DIM) return;
    const int n = i >> 9;              // /512
    const int h = (i >> 6) & 7;        // (i%512)/64
    agg1[i] = elu(agg1[i] / (rowsum1[n * 8 + h] + EPS));
}

// out <- elu(agg2 / (rowsum2 + eps))
__global__ __launch_bounds__(256) void k_final_out(const float* __restrict__ agg2,
                                                   const float* __restrict__ rowsum2,
                                                   float* __restrict__ out) {
    const int i = blockIdx.x * 256 + threadIdx.x;
    if (i >= N_NODES * EMBED_DIM) return;
    out[i] = elu(agg2[i] / (rowsum2[i >> 7] + EPS));
}

// copy edge_index (bitwise) + attn_out = e / (rowsum2[src] + eps)
__global__ __launch_bounds__(256) void k_final_edges(const int* __restrict__ eidx,
                                                     const float* __restrict__ ework,
                                                     const float* __restrict__ rowsum2,
                                                     const int* __restrict__ src,
                                                     int* __restrict__ out_eidx,
                                                     float* __restrict__ out_attn) {
    const int i = blockIdx.x * 256 + threadIdx.x;
    if (i < 2 * N_EDGES) out_eidx[i] = eidx[i];
    if (i < N_EDGES)     out_attn[i] = ework[i] / (rowsum2[src[i]] + EPS);
}

extern "C" void kernel_launch(void* const* d_in, const int* in_sizes, int n_in,
                              void* d_out, int out_size, void* d_ws, size_t ws_size,
                              hipStream_t stream) {
    (void)in_sizes; (void)n_in; (void)out_size; (void)ws_size;
    const float* x     = (const float*)d_in[0];   // [50000,256]
    const float* W_in  = (const float*)d_in[1];   // [8,256,64]
    const float* a_in  = (const float*)d_in[2];   // [8,128]
    const float* W_out = (const float*)d_in[3];   // [512,128]
    const float* a_out = (const float*)d_in[4];   // [256]
    const int*   eidx  = (const int*)d_in[5];     // [2,1600000]
    const int* src = eidx;
    const int* dst = eidx + N_EDGES;

    // workspace carve-up (floats)
    float* ws = (float*)d_ws;
    float* h_all = ws;  ws += (size_t)N_NODES * CAT_DIM;     // 25.6M
    float* agg1  = ws;  ws += (size_t)N_NODES * CAT_DIM;     // 25.6M (becomes xcat)
    float* as1   = ws;  ws += (size_t)N_NODES * NHEADS;
    float* ad1   = ws;  ws += (size_t)N_NODES * NHEADS;
    float* rs1   = ws;  ws += (size_t)N_NODES * NHEADS;
    float* h2    = ws;  ws += (size_t)N_NODES * EMBED_DIM;   // 6.4M
    float* agg2  = ws;  ws += (size_t)N_NODES * EMBED_DIM;   // 6.4M
    float* as2   = ws;  ws += (size_t)N_NODES;
    float* ad2   = ws;  ws += (size_t)N_NODES;
    float* rs2   = ws;  ws += (size_t)N_NODES;
    float* ework = ws;  ws += (size_t)N_EDGES;

    // output layout: out[50000*128] ++ edge_index[2*1.6M] ++ attn[1.6M]
    float* out_h    = (float*)d_out;
    int*   out_eidx = (int*)((float*)d_out + (size_t)N_NODES * EMBED_DIM);
    float* out_attn = (float*)d_out + (size_t)N_NODES * EMBED_DIM + 2 * (size_t)N_EDGES;

    const dim3 b(256);
    // zero accumulators (required every call: atomically accumulated)
    k_zero<<<(N_NODES * CAT_DIM + 255) / 256, b, 0, stream>>>(agg1, N_NODES * CAT_DIM);
    k_zero<<<(N_NODES * NHEADS + 255) / 256, b, 0, stream>>>(rs1, N_NODES * NHEADS);
    k_zero<<<(N_NODES * EMBED_DIM + 255) / 256, b, 0, stream>>>(agg2, N_NODES * EMBED_DIM);
    k_zero<<<(N_NODES + 255) / 256, b, 0, stream>>>(rs2, N_NODES);

    // ---- layer 1 ----
    k_wmma_gemm<0><<<(N_NODES / 16) * (CAT_DIM / 128), b, 0, stream>>>(x, W_in, h_all,
                                                                       INIT_DIM, CAT_DIM);
    k_node_scalars1<<<N_NODES, b, 0, stream>>>(h_all, a_in, as1, ad1);  // 400k waves
    k_edge1<<<(N_EDGES + 7) / 8, b, 0, stream>>>(src, dst, h_all, as1, ad1, rs1, agg1);
    k_norm_elu1<<<(N_NODES * CAT_DIM + 255) / 256, b, 0, stream>>>(agg1, rs1);

    // ---- layer 2 ----
    k_wmma_gemm<1><<<(N_NODES / 16) * (EMBED_DIM / 128), b, 0, stream>>>(agg1, W_out, h2,
                                                                         CAT_DIM, EMBED_DIM);
    k_node_scalars2<<<(N_NODES + 7) / 8, b, 0, stream>>>(h2, a_out, as2, ad2);
    k_edge2<<<(N_EDGES + 7) / 8, b, 0, stream>>>(src, dst, h2, as2, ad2, rs2, agg2, ework);

    // ---- outputs ----
    k_final_out<<<(N_NODES * EMBED_DIM + 255) / 256, b, 0, stream>>>(agg2, rs2, out_h);
    k_final_edges<<<(2 * N_EDGES + 255) / 256, b, 0, stream>>>(eidx, ework, rs2, src,
                                                               out_eidx, out_attn);
}